// EnhancedGraphBlock_84061099917997
// MI455X (gfx1250) — compile-verified
//
#include <hip/hip_runtime.h>
#include <hip/hip_bf16.h>
#include <math.h>

// ---------------------------------------------------------------- constants
#define N_NODES 20000
#define E_EDGES 320000
#define ETOT    (E_EDGES + N_NODES)   // with self-loops
#define F_IN    128
#define HEADS   4
#define CH      64
#define HC      256                   // HEADS*CH
#define NGRAPH  64
#define BN_EPS  1e-5f

typedef float v2f __attribute__((ext_vector_type(2)));
typedef float v8f __attribute__((ext_vector_type(8)));

// ---------------------------------------------------------------- helpers
__device__ __forceinline__ int edge_src(const int* __restrict__ ei, int i) {
    return (i < E_EDGES) ? ei[i] : (i - E_EDGES);
}
__device__ __forceinline__ int edge_dst(const int* __restrict__ ei, int i) {
    return (i < E_EDGES) ? ei[E_EDGES + i] : (i - E_EDGES);
}
__device__ __forceinline__ void atomicMaxF(float* addr, float v) {
    // classic signed/unsigned split: correct total order on float bits
    if (v >= 0.0f) atomicMax((int*)addr, __float_as_int(v));
    else           atomicMin((unsigned int*)addr, __float_as_uint(v));
}
__device__ __forceinline__ float gelu_exact(float x) {
    return 0.5f * x * (1.0f + erff(x * 0.70710678118654752f));
}

// ---------------------------------------------------------------- fill
__global__ void fill_kernel(float* __restrict__ p, float v, int n) {
    int i = blockIdx.x * blockDim.x + threadIdx.x;
    if (i < n) p[i] = v;
}

// ---------------------------------------------------------------- fp32 WMMA GEMM
// C[M,Nc] = A[M,K] @ B[K,Nc], row-major, fp32 end to end via
// V_WMMA_F32_16X16X4_F32.  One wave -> one 16x16 C tile, 4 waves/block ->
// 16x64 strip.  M%16==0, K%4==0, Nc%64==0 for all call sites here.
__global__ void __launch_bounds__(128)
gemm_wmma_f32(const float* __restrict__ A, const float* __restrict__ B,
              float* __restrict__ C, int M, int K, int Nc) {
    const int lane = threadIdx.x & 31;
    const int wave = threadIdx.x >> 5;
    const int row0 = blockIdx.x << 4;
    const int col0 = (blockIdx.y << 6) + (wave << 4);
    const int m    = lane & 15;            // row (A) / col (B,C) within tile
    const int kh   = (lane >> 4) << 1;     // K sub-offset: 0 (lanes 0-15), 2 (16-31)

    v8f acc = {};
    const float* arow = A + (size_t)(row0 + m) * K;
    for (int k = 0; k < K; k += 4) {
        v2f a, b;
        a.x = arow[k + kh];
        a.y = arow[k + kh + 1];
        b.x = B[(size_t)(k + kh)     * Nc + col0 + m];
        b.y = B[(size_t)(k + kh + 1) * Nc + col0 + m];
        acc = __builtin_amdgcn_wmma_f32_16x16x4_f32(
                  /*neg_a=*/false, a, /*neg_b=*/false, b,
                  /*c_mod=*/(short)0, acc, /*reuse_a=*/false, /*reuse_b=*/false);
    }
    const int half = lane >> 4;            // C/D layout: M split 0-7 / 8-15
#pragma unroll
    for (int r = 0; r < 8; ++r)
        C[(size_t)(row0 + half * 8 + r) * Nc + col0 + m] = acc[r];
}

// ---------------------------------------------------------------- attention coefs
// es[n,h] = <hproj[n, h*64 : h*64+64], a_src[h]>, likewise ed.
__global__ void attn_coef_kernel(const float* __restrict__ hproj,
                                 const float* __restrict__ a_src,
                                 const float* __restrict__ a_dst,
                                 float* __restrict__ es, float* __restrict__ ed) {
    int idx = blockIdx.x * blockDim.x + threadIdx.x;
    if (idx >= N_NODES * HEADS) return;
    int n = idx >> 2, h = idx & 3;
    const float* hp = hproj + (size_t)n * HC + h * CH;
    const float* as = a_src + h * CH;
    const float* ad = a_dst + h * CH;
    float s = 0.0f, d = 0.0f;
#pragma unroll 4
    for (int c = 0; c < CH; ++c) { s += hp[c] * as[c]; d += hp[c] * ad[c]; }
    es[idx] = s; ed[idx] = d;
}

// ---------------------------------------------------------------- edge pass 1: segment max
__global__ void edge_max_kernel(const int* __restrict__ ei,
                                const float* __restrict__ es,
                                const float* __restrict__ ed,
                                float* __restrict__ maxb, int etot) {
    int idx = blockIdx.x * blockDim.x + threadIdx.x;
    if (idx >= etot * HEADS) return;
    int i = idx >> 2, h = idx & 3;
    int s = edge_src(ei, i), d = edge_dst(ei, i);
    float t = es[s * HEADS + h] + ed[d * HEADS + h];
    float e = t > 0.0f ? t : 0.2f * t;                 // leaky_relu(0.2)
    atomicMaxF(&maxb[d * HEADS + h], e);
}

// ---------------------------------------------------------------- edge pass 2: segment sum of exp
__global__ void edge_sum_kernel(const int* __restrict__ ei,
                                const float* __restrict__ es,
                                const float* __restrict__ ed,
                                const float* __restrict__ maxb,
                                float* __restrict__ sumb, int etot) {
    int idx = blockIdx.x * blockDim.x + threadIdx.x;
    if (idx >= etot * HEADS) return;
    int i = idx >> 2, h = idx & 3;
    int s = edge_src(ei, i), d = edge_dst(ei, i);
    float t = es[s * HEADS + h] + ed[d * HEADS + h];
    float e = t > 0.0f ? t : 0.2f * t;
    float p = expf(e - maxb[d * HEADS + h]);
    atomicAdd(&sumb[d * HEADS + h], p);
}

// ---------------------------------------------------------------- edge pass 3: weighted scatter
// agg[d,c] += (1/H) * sum_h hproj[s, h*64+c] * alpha[h]   (head-mean folded in)
__global__ void __launch_bounds__(256)
edge_scatter_kernel(const int* __restrict__ ei,
                    const float* __restrict__ es,
                    const float* __restrict__ ed,
                    const float* __restrict__ maxb,
                    const float* __restrict__ sumb,
                    const float* __restrict__ hproj,
                    float* __restrict__ agg, int etot) {
    __shared__ float alpha[4][HEADS];
    int tx = threadIdx.x;                  // channel 0..63
    int ty = threadIdx.y;                  // local edge 0..3
    int eidx = blockIdx.x * 4 + ty;
    bool valid = eidx < etot;
    int s = 0, d = 0;
    if (valid) { s = edge_src(ei, eidx); d = edge_dst(ei, eidx); }
    if (valid && tx < HEADS) {
        int h = tx;
        float t = es[s * HEADS + h] + ed[d * HEADS + h];
        float e = t > 0.0f ? t : 0.2f * t;
        float p = expf(e - maxb[d * HEADS + h]);
        alpha[ty][h] = p / (sumb[d * HEADS + h] + 1e-16f);
    }
    __syncthreads();
    if (valid) {
        const float* hp = hproj + (size_t)s * HC;
        float v = 0.25f * (hp[tx]           * alpha[ty][0] +
                           hp[CH + tx]      * alpha[ty][1] +
                           hp[2 * CH + tx]  * alpha[ty][2] +
                           hp[3 * CH + tx]  * alpha[ty][3]);
        atomicAdd(&agg[(size_t)d * CH + tx], v);
    }
}

// ---------------------------------------------------------------- BN: partial sums
// sums[0:64] = sum_c, sums[64:128] = sumsq_c over (agg[n,c] + bias[c])
__global__ void __launch_bounds__(256)
bn_reduce_kernel(const float* __restrict__ agg, const float* __restrict__ bias,
                 float* __restrict__ sums) {
    __shared__ float s1[256], s2[256];
    int tid = threadIdx.x;
    int c = tid & 63, rg = tid >> 6;       // 4 row groups
    float a = 0.0f, a2 = 0.0f;
    float b = bias[c];
    for (int n = blockIdx.x * 4 + rg; n < N_NODES; n += gridDim.x * 4) {
        float v = agg[(size_t)n * CH + c] + b;
        a += v; a2 += v * v;
    }
    s1[tid] = a; s2[tid] = a2;
    __syncthreads();
    if (tid < 64) {
        float t1 = s1[c] + s1[64 + c] + s1[128 + c] + s1[192 + c];
        float t2 = s2[c] + s2[64 + c] + s2[128 + c] + s2[192 + c];
        atomicAdd(&sums[c], t1);
        atomicAdd(&sums[64 + c], t2);
    }
}

// stats[0:64]=mean, stats[64:128]=rstd
__global__ void bn_finalize_kernel(const float* __restrict__ sums,
                                   float* __restrict__ stats) {
    int c = threadIdx.x;
    if (c >= 64) return;
    float mean = sums[c] / (float)N_NODES;
    float var  = sums[64 + c] / (float)N_NODES - mean * mean;
    stats[c] = mean;
    stats[64 + c] = rsqrtf(var + BN_EPS);
}

// out[n,c] = gelu( bn(agg+bias) + extra[n,c] + (has_eb ? eb[c] : 0) )
__global__ void bn_apply_kernel(const float* __restrict__ agg,
                                const float* __restrict__ bias,
                                const float* __restrict__ stats,
                                const float* __restrict__ gamma,
                                const float* __restrict__ beta,
                                const float* __restrict__ extra,
                                const float* __restrict__ eb, int has_eb,
                                float* __restrict__ outp) {
    int idx = blockIdx.x * blockDim.x + threadIdx.x;
    if (idx >= N_NODES * CH) return;
    int c = idx & 63;
    float v  = agg[idx] + bias[c];
    float xn = (v - stats[c]) * stats[64 + c] * gamma[c] + beta[c];
    float ex = extra[idx] + (has_eb ? eb[c] : 0.0f);
    outp[idx] = gelu_exact(xn + ex);
}

// ---------------------------------------------------------------- pooling
__global__ void pool_accum_kernel(const float* __restrict__ h,
                                  const int* __restrict__ batch,
                                  float* __restrict__ pool) {
    int idx = blockIdx.x * blockDim.x + threadIdx.x;
    if (idx >= N_NODES * CH) return;
    int n = idx >> 6, c = idx & 63;
    atomicAdd(&pool[batch[n] * CH + c], h[idx]);
}
__global__ void count_kernel(const int* __restrict__ batch, float* __restrict__ cnt) {
    int n = blockIdx.x * blockDim.x + threadIdx.x;
    if (n < N_NODES) atomicAdd(&cnt[batch[n]], 1.0f);
}
__global__ void pool_final_kernel(const float* __restrict__ pool,
                                  const float* __restrict__ cnt,
                                  float* __restrict__ out) {
    int idx = blockIdx.x * blockDim.x + threadIdx.x;
    if (idx >= NGRAPH * CH) return;
    out[idx] = pool[idx] / fmaxf(cnt[idx >> 6], 1.0f);
}

// ---------------------------------------------------------------- workspace map (floats)
#define OFF_HPROJ 0u                               // N*256  (reused both layers)
#define OFF_ES    (OFF_HPROJ + N_NODES * HC)       // N*4
#define OFF_ED    (OFF_ES    + N_NODES * HEADS)
#define OFF_MAX   (OFF_ED    + N_NODES * HEADS)
#define OFF_SUM   (OFF_MAX   + N_NODES * HEADS)
#define OFF_AGG   (OFF_SUM   + N_NODES * HEADS)    // N*64 (reused both layers)
#define OFF_SKIP  (OFF_AGG   + N_NODES * CH)       // N*64 (skip, later h_final)
#define OFF_HMID  (OFF_SKIP  + N_NODES * CH)       // N*64
#define OFF_SUMS  (OFF_HMID  + N_NODES * CH)       // 128
#define OFF_STATS (OFF_SUMS  + 128)                // 128
#define OFF_POOL  (OFF_STATS + 128)                // 64*64
#define OFF_CNT   (OFF_POOL  + NGRAPH * CH)        // 64

extern "C" void kernel_launch(void* const* d_in, const int* in_sizes, int n_in,
                              void* d_out, int out_size, void* d_ws, size_t ws_size,
                              hipStream_t stream) {
    const float* x      = (const float*)d_in[0];
    const float* W1     = (const float*)d_in[1];
    const float* a_src1 = (const float*)d_in[2];
    const float* a_dst1 = (const float*)d_in[3];
    const float* b1     = (const float*)d_in[4];
    const float* Wskip  = (const float*)d_in[5];
    const float* bskip  = (const float*)d_in[6];
    const float* g1     = (const float*)d_in[7];
    const float* be1    = (const float*)d_in[8];
    const float* W2     = (const float*)d_in[9];
    const float* a_src2 = (const float*)d_in[10];
    const float* a_dst2 = (const float*)d_in[11];
    const float* b2     = (const float*)d_in[12];
    const float* g2     = (const float*)d_in[13];
    const float* be2    = (const float*)d_in[14];
    const int*   ei     = (const int*)d_in[15];
    const int*   batch  = (const int*)d_in[16];
    float* out = (float*)d_out;

    float* ws    = (float*)d_ws;
    float* hproj = ws + OFF_HPROJ;
    float* es    = ws + OFF_ES;
    float* ed    = ws + OFF_ED;
    float* maxb  = ws + OFF_MAX;
    float* sumb  = ws + OFF_SUM;
    float* agg   = ws + OFF_AGG;
    float* skipb = ws + OFF_SKIP;   // layer-1 skip; reused as h_final
    float* hmid  = ws + OFF_HMID;
    float* sums  = ws + OFF_SUMS;
    float* stats = ws + OFF_STATS;
    float* pool  = ws + OFF_POOL;
    float* cnt   = ws + OFF_CNT;

    const int FT = 256;
    auto cdiv = [](int a, int b) { return (a + b - 1) / b; };

    // ================= layer 1 =================
    gemm_wmma_f32<<<dim3(N_NODES / 16, HC / 64), 128, 0, stream>>>(x, W1, hproj, N_NODES, F_IN, HC);
    attn_coef_kernel<<<cdiv(N_NODES * HEADS, FT), FT, 0, stream>>>(hproj, a_src1, a_dst1, es, ed);

    fill_kernel<<<cdiv(N_NODES * HEADS, FT), FT, 0, stream>>>(maxb, -INFINITY, N_NODES * HEADS);
    fill_kernel<<<cdiv(N_NODES * HEADS, FT), FT, 0, stream>>>(sumb, 0.0f, N_NODES * HEADS);
    fill_kernel<<<cdiv(N_NODES * CH, FT), FT, 0, stream>>>(agg, 0.0f, N_NODES * CH);

    edge_max_kernel<<<cdiv(ETOT * HEADS, FT), FT, 0, stream>>>(ei, es, ed, maxb, ETOT);
    edge_sum_kernel<<<cdiv(ETOT * HEADS, FT), FT, 0, stream>>>(ei, es, ed, maxb, sumb, ETOT);
    edge_scatter_kernel<<<cdiv(ETOT, 4), dim3(64, 4), 0, stream>>>(ei, es, ed, maxb, sumb, hproj, agg, ETOT);

    gemm_wmma_f32<<<dim3(N_NODES / 16, 1), 128, 0, stream>>>(x, Wskip, skipb, N_NODES, F_IN, CH);

    fill_kernel<<<1, 128, 0, stream>>>(sums, 0.0f, 128);
    bn_reduce_kernel<<<80, 256, 0, stream>>>(agg, b1, sums);
    bn_finalize_kernel<<<1, 64, 0, stream>>>(sums, stats);
    bn_apply_kernel<<<cdiv(N_NODES * CH, FT), FT, 0, stream>>>(agg, b1, stats, g1, be1,
                                                               skipb, bskip, 1, hmid);

    // ================= layer 2 =================
    gemm_wmma_f32<<<dim3(N_NODES / 16, HC / 64), 128, 0, stream>>>(hmid, W2, hproj, N_NODES, CH, HC);
    attn_coef_kernel<<<cdiv(N_NODES * HEADS, FT), FT, 0, stream>>>(hproj, a_src2, a_dst2, es, ed);

    fill_kernel<<<cdiv(N_NODES * HEADS, FT), FT, 0, stream>>>(maxb, -INFINITY, N_NODES * HEADS);
    fill_kernel<<<cdiv(N_NODES * HEADS, FT), FT, 0, stream>>>(sumb, 0.0f, N_NODES * HEADS);
    fill_kernel<<<cdiv(N_NODES * CH, FT), FT, 0, stream>>>(agg, 0.0f, N_NODES * CH);

    edge_max_kernel<<<cdiv(ETOT * HEADS, FT), FT, 0, stream>>>(ei, es, ed, maxb, ETOT);
    edge_sum_kernel<<<cdiv(ETOT * HEADS, FT), FT, 0, stream>>>(ei, es, ed, maxb, sumb, ETOT);
    edge_scatter_kernel<<<cdiv(ETOT, 4), dim3(64, 4), 0, stream>>>(ei, es, ed, maxb, sumb, hproj, agg, ETOT);

    fill_kernel<<<1, 128, 0, stream>>>(sums, 0.0f, 128);
    bn_reduce_kernel<<<80, 256, 0, stream>>>(agg, b2, sums);
    bn_finalize_kernel<<<1, 64, 0, stream>>>(sums, stats);
    // residual is hmid; no extra bias. h_final reuses skipb.
    bn_apply_kernel<<<cdiv(N_NODES * CH, FT), FT, 0, stream>>>(agg, b2, stats, g2, be2,
                                                               hmid, nullptr, 0, skipb);

    // ================= global mean pool =================
    fill_kernel<<<cdiv(NGRAPH * CH, FT), FT, 0, stream>>>(pool, 0.0f, NGRAPH * CH);
    fill_kernel<<<1, 64, 0, stream>>>(cnt, 0.0f, NGRAPH);
    pool_accum_kernel<<<cdiv(N_NODES * CH, FT), FT, 0, stream>>>(skipb, batch, pool);
    count_kernel<<<cdiv(N_NODES, FT), FT, 0, stream>>>(batch, cnt);
    pool_final_kernel<<<cdiv(NGRAPH * CH, FT), FT, 0, stream>>>(pool, cnt, out);
}